// Detection_13056700580348
// MI455X (gfx1250) — compile-verified
//
#include <hip/hip_runtime.h>

typedef float v2f __attribute__((ext_vector_type(2)));
typedef float v8f __attribute__((ext_vector_type(8)));

#define S_GRID   56
#define N_CLASS  20
#define CELLS    (S_GRID * S_GRID)          // 3136
#define ITEMS    (CELLS * 2)                // 6272
#define SORTN    8192
#define CAND     256
#define MAXOUT   30
#define D_IMG    94080
#define CONF_OFF 62720                      // 20*3136
#define COORD_OFF 68992                     // 62720 + 2*3136
#define NEG_INF  (-3.402823466e38f)

struct Phase2 {
    float y0[CAND], x0[CAND], y1[CAND], x1[CAND];
    float sc[CAND], cl[CAND], ar[CAND];
    float redv[CAND];
    int   redi[CAND];
    unsigned short mask16[CAND * 16];       // row-major: [row][col16]
    int selIdx[32], selOk[32];
    unsigned char valid[CAND];
};

__global__ __launch_bounds__(256) void det_kernel(const float* __restrict__ net,
                                                  float* __restrict__ out) {
    __shared__ unsigned long long keys[SORTN];              // 64 KB
    Phase2* P = reinterpret_cast<Phase2*>(&keys[CAND]);     // carve tail (~18 KB of 63.5 KB)

    const int tid = threadIdx.x;
    const int img = blockIdx.x;
    const float* in = net + (size_t)img * D_IMG;

    // ---------------- Phase 1: scores -> sort keys ----------------
    for (int cell = tid; cell < CELLS; cell += 256) {
        const float* pr = in + cell * N_CLASS;
        float p[N_CLASS];
        #pragma unroll
        for (int c = 0; c < N_CLASS; ++c) p[c] = pr[c];
        #pragma unroll
        for (int b = 0; b < 2; ++b) {
            int id = cell * 2 + b;
            float conf = in[CONF_OFF + id];
            float best = NEG_INF;
            #pragma unroll
            for (int c = 0; c < N_CLASS; ++c) {
                float v = __fmul_rn(conf, p[c]);
                best = v > best ? v : best;
            }
            float s = (best >= 0.1f) ? best : 0.0f;
            keys[id] = ((unsigned long long)__float_as_uint(s) << 32) |
                       (unsigned)(~(unsigned)id);
        }
    }
    for (int q = ITEMS + tid; q < SORTN; q += 256)
        keys[q] = (unsigned long long)(unsigned)(~(unsigned)q);   // below all real keys
    __syncthreads();

    // ---------------- Phase 1b: bitonic sort, descending ----------------
    for (unsigned k = 2; k <= (unsigned)SORTN; k <<= 1) {
        for (unsigned j = k >> 1; j > 0; j >>= 1) {
            for (unsigned i = tid; i < (unsigned)SORTN; i += 256u) {
                unsigned x = i ^ j;
                if (x > i) {
                    unsigned long long a = keys[i], b = keys[x];
                    bool up = ((i & k) == 0u);
                    if (up ? (a < b) : (a > b)) { keys[i] = b; keys[x] = a; }
                }
            }
            __syncthreads();
        }
    }

    // ---------------- Phase 2: decode top-256 candidates ----------------
    {
        unsigned long long key = keys[tid];
        unsigned id = ~(unsigned)key;                 // item index
        float s = __uint_as_float((unsigned)(key >> 32));
        int cell = (int)(id >> 1);
        int gi = cell / S_GRID, gj = cell % S_GRID;
        const float* cd = in + COORD_OFF + (size_t)id * 4;
        float tx = cd[0], ty = cd[1], tw = cd[2], th = cd[3];
        float x = __fdiv_rn(__fadd_rn(tx, (float)gj), 56.0f);
        float y = __fdiv_rn(__fadd_rn(ty, (float)gi), 56.0f);
        float hw = __fmul_rn(__fmul_rn(tw, tw), 0.5f);
        float hh = __fmul_rn(__fmul_rn(th, th), 0.5f);
        float y0 = __fsub_rn(y, hh), x0 = __fsub_rn(x, hw);
        float y1 = __fadd_rn(y, hh), x1 = __fadd_rn(x, hw);
        float conf = in[CONF_OFF + id];
        const float* pr = in + cell * N_CLASS;
        float best = NEG_INF; int bc = 0;
        #pragma unroll
        for (int c = 0; c < N_CLASS; ++c) {
            float v = __fmul_rn(conf, pr[c]);
            if (v > best) { best = v; bc = c; }       // first-max == jnp.argmax
        }
        float dy = fmaxf(__fsub_rn(y1, y0), 0.0f);
        float dx = fmaxf(__fsub_rn(x1, x0), 0.0f);
        P->y0[tid] = y0; P->x0[tid] = x0; P->y1[tid] = y1; P->x1[tid] = x1;
        P->sc[tid] = s;  P->cl[tid] = (float)bc;
        P->ar[tid] = __fmul_rn(dy, dx);
        P->valid[tid] = 1;
    }
    __syncthreads();

    // ------- Phase 3: WMMA area-sum + IoU suppression mask (256x256) -------
    {
        const int lane = tid & 31;
        const int wave = tid >> 5;                    // 8 waves, 32 tiles each
        v8f czero = {};
        for (int tt = wave * 32; tt < wave * 32 + 32; ++tt) {
            int mi = tt >> 4, ni = tt & 15;
            int m0 = mi << 4, n0 = ni << 4;
            // A (16x4): [area_m, 1, 0, 0]; B (4x16): [1; area_n; 0; 0]
            v2f A, B;
            if (lane < 16) {
                A[0] = P->ar[m0 + lane]; A[1] = 1.0f;
                B[0] = 1.0f;             B[1] = P->ar[n0 + lane];
            } else {
                A[0] = 0.0f; A[1] = 0.0f; B[0] = 0.0f; B[1] = 0.0f;
            }
            v8f d = __builtin_amdgcn_wmma_f32_16x16x4_f32(
                false, A, false, B, (short)0, czero, false, false);
            // D layout: VGPR r -> rows (m0+r, m0+8+r); lane%16 -> column
            int n = n0 + (lane & 15);
            float ny0 = P->y0[n], nx0 = P->x0[n], ny1 = P->y1[n], nx1 = P->x1[n];
            int mbase = m0 + ((lane & 16) ? 8 : 0);
            #pragma unroll
            for (int r = 0; r < 8; ++r) {
                int m = mbase + r;
                float iy = fmaxf(__fsub_rn(fminf(P->y1[m], ny1), fmaxf(P->y0[m], ny0)), 0.0f);
                float ix = fmaxf(__fsub_rn(fminf(P->x1[m], nx1), fmaxf(P->x0[m], nx0)), 0.0f);
                float inter = __fmul_rn(iy, ix);
                float uni = __fsub_rn(d[r], inter);   // (area_m + area_n) - inter
                bool pred = (uni > 0.0f) && (__fdiv_rn(inter, uni) > 0.4f);
                unsigned bal = __builtin_amdgcn_ballot_w32(pred);
                if (lane == 0) {
                    P->mask16[(m0 + r)     * 16 + ni] = (unsigned short)(bal & 0xFFFFu);
                    P->mask16[(m0 + 8 + r) * 16 + ni] = (unsigned short)(bal >> 16);
                }
            }
        }
    }
    __syncthreads();

    // ---------------- Phase 4: greedy NMS scan (30 steps) ----------------
    for (int it = 0; it < MAXOUT; ++it) {
        P->redv[tid] = P->valid[tid] ? P->sc[tid] : NEG_INF;
        P->redi[tid] = tid;
        __syncthreads();
        for (int off = 128; off > 0; off >>= 1) {
            if (tid < off) {
                float v2 = P->redv[tid + off]; int i2 = P->redi[tid + off];
                if (v2 > P->redv[tid] || (v2 == P->redv[tid] && i2 < P->redi[tid])) {
                    P->redv[tid] = v2; P->redi[tid] = i2;
                }
            }
            __syncthreads();
        }
        int idx = P->redi[0];
        int ok  = P->valid[idx];
        unsigned bit = (P->mask16[idx * 16 + (tid >> 4)] >> (tid & 15)) & 1u;
        unsigned char nv = (unsigned char)(P->valid[tid] & (unsigned char)(1u - bit));
        if (tid == idx) nv = 0;
        __syncthreads();                   // all reads of valid[] done before writes
        P->valid[tid] = nv;
        if (tid == 0) { P->selIdx[it] = idx; P->selOk[it] = ok; }
        __syncthreads();
    }

    // ---------------- Phase 5: rank the 30 rows, write output ----------------
    if (tid < MAXOUT) {
        int sel = P->selIdx[tid];
        float okf = P->selOk[tid] ? 1.0f : 0.0f;
        P->redv[tid] = __fmul_rn(P->sc[sel], okf);
    }
    __syncthreads();
    if (tid < MAXOUT) {
        int sel = P->selIdx[tid];
        float okf = P->selOk[tid] ? 1.0f : 0.0f;
        float ms = P->redv[tid];
        int rank = 0;
        for (int jt = 0; jt < MAXOUT; ++jt) {
            float vj = P->redv[jt];
            if (vj > ms || (vj == ms && jt < tid)) rank++;   // stable top_k
        }
        float* o = out + (size_t)img * (MAXOUT * 6) + (size_t)rank * 6;
        o[0] = __fmul_rn(P->y0[sel], okf);
        o[1] = __fmul_rn(P->x0[sel], okf);
        o[2] = __fmul_rn(P->y1[sel], okf);
        o[3] = __fmul_rn(P->x1[sel], okf);
        o[4] = ms;
        o[5] = __fmul_rn(P->cl[sel], okf);
    }
}

extern "C" void kernel_launch(void* const* d_in, const int* in_sizes, int n_in,
                              void* d_out, int out_size, void* d_ws, size_t ws_size,
                              hipStream_t stream) {
    (void)n_in; (void)out_size; (void)d_ws; (void)ws_size;
    const float* net = (const float*)d_in[0];
    float* out = (float*)d_out;
    int batch = in_sizes[0] / D_IMG;        // 256
    det_kernel<<<dim3(batch), dim3(256), 0, stream>>>(net, out);
}